// GNNvirtual_37495064494619
// MI455X (gfx1250) — compile-verified
//
#include <hip/hip_runtime.h>
#include <hip/hip_bf16.h>
#include <float.h>
#include <math.h>

#define NN 102400
#define NE 1638400
#define NG 256
#define NH 64
#define GDEPTH 3
#define LN_EPS 1e-5f
#define BN_EPS 1e-5f

typedef __attribute__((ext_vector_type(2))) float v2f;
typedef __attribute__((ext_vector_type(4))) float v4f;
typedef __attribute__((ext_vector_type(8))) float v8f;

__device__ __forceinline__ float wsum32(float v) {
#pragma unroll
  for (int o = 16; o > 0; o >>= 1) v += __shfl_xor(v, o, 32);
  return v;
}

__device__ __forceinline__ void atomicMaxF(float* addr, float val) {
  if (val >= 0.f) atomicMax((int*)addr, __float_as_int(val));
  else            atomicMin((unsigned int*)addr, __float_as_uint(val));
}

// ---------------------------------------------------------------------------
// Generic fp32 WMMA GEMM:  Out[M,NOUT] = produce_A(M,64) @ W[64,NOUT] + bias
// MODE 0: A from buffer.
// MODE 1: A[m][k] = relu(cfg[m]*w1[k] + b1[k])                (node MLP l1 fused)
// MODE 2: A[m][k] = relu(deg[r[m]]*w1[k]+deg[c[m]]*w1[64+k]+b1[k]) (edge l1 fused)
// MODE 3: A[m][k] = alpha*A[m][k] + P1[m][k], alpha = 1 + *epsp (GINE input fused)
// NTST: non-temporal stores (for the 419MB edge_attr stream -> keep L2 clean).
// One block = 64 rows; waves tile (4 row-tiles) x (NOUT/16 col-tiles).
// K=64 -> 16 x V_WMMA_F32_16X16X4_F32 per wave. Weights staged in LDS.
// ---------------------------------------------------------------------------
template <int MODE, int NOUT, bool RELU, bool NTST>
__global__ __launch_bounds__(32 * 4 * (NOUT / 16))
void gemm_wmma(const float* __restrict__ A, const float* __restrict__ P1,
               const int* __restrict__ I0, const int* __restrict__ I1,
               const float* __restrict__ Wm, const float* __restrict__ Bias,
               const float* __restrict__ w1p, const float* __restrict__ b1p,
               const float* __restrict__ epsp, float* __restrict__ Out) {
  __shared__ float Wl[64 * NOUT];
  __shared__ float bl[NOUT];
  __shared__ float w1l[128];
  __shared__ float b1l[64];
  const int tid = threadIdx.x;
  const int nth = 32 * 4 * (NOUT / 16);
  for (int i = tid; i < 64 * NOUT; i += nth) Wl[i] = Wm[i];
  for (int i = tid; i < NOUT; i += nth) bl[i] = Bias[i];
  if (MODE == 1 || MODE == 2) {
    for (int i = tid; i < (MODE == 2 ? 128 : 64); i += nth) w1l[i] = w1p[i];
    for (int i = tid; i < 64; i += nth) b1l[i] = b1p[i];
  }
  __syncthreads();

  const int lane = tid & 31;
  const int wv = tid >> 5;
  const int CT = NOUT / 16;
  const int rt = wv / CT, ct = wv % CT;
  const long m0 = (long)blockIdx.x * 64 + rt * 16;
  const long mrow = m0 + (lane & 15);
  const int khalf = lane >> 4;
  const int ncol = ct * 16 + (lane & 15);

  float s0 = 0.f, s1 = 0.f, alpha = 1.f;
  if (MODE == 1) s0 = P1[mrow];
  if (MODE == 2) { s0 = P1[I0[mrow]]; s1 = P1[I1[mrow]]; }
  if (MODE == 3 && epsp != nullptr) alpha = 1.f + *epsp;

  v8f acc = {};
#pragma unroll
  for (int k0 = 0; k0 < 64; k0 += 4) {
    const int k = k0 + khalf * 2;  // lane's K pair within the 16x4 A fragment
    v2f a;
    if (MODE == 0) {
      const float2 t = *(const float2*)(A + mrow * 64 + k);
      a.x = t.x; a.y = t.y;
    } else if (MODE == 1) {
      a.x = fmaxf(fmaf(s0, w1l[k], b1l[k]), 0.f);
      a.y = fmaxf(fmaf(s0, w1l[k + 1], b1l[k + 1]), 0.f);
    } else if (MODE == 2) {
      a.x = fmaxf(s0 * w1l[k] + s1 * w1l[64 + k] + b1l[k], 0.f);
      a.y = fmaxf(s0 * w1l[k + 1] + s1 * w1l[64 + k + 1] + b1l[k + 1], 0.f);
    } else {
      const long off = mrow * 64 + k;
      a.x = fmaf(alpha, A[off], P1[off]);
      a.y = fmaf(alpha, A[off + 1], P1[off + 1]);
    }
    v2f b;
    b.x = Wl[k * NOUT + ncol];
    b.y = Wl[(k + 1) * NOUT + ncol];
    acc = __builtin_amdgcn_wmma_f32_16x16x4_f32(false, a, false, b, (short)0,
                                                acc, false, false);
  }

  const long crow = m0 + (long)khalf * 8;  // C: M = v + 8*(lane>=16)
  const float bb = bl[ncol];
#pragma unroll
  for (int v = 0; v < 8; v++) {
    float val = acc[v] + bb;
    if (RELU) val = fmaxf(val, 0.f);
    float* p = Out + (crow + v) * NOUT + ncol;
    if (NTST) __builtin_nontemporal_store(val, p);
    else      *p = val;
  }
}

// ---------------------------------------------------------------------------
__global__ void k_deg(const int* __restrict__ col, float* __restrict__ deg) {
  int e = blockIdx.x * blockDim.x + threadIdx.x;
  if (e < NE) atomicAdd(&deg[col[e]], 1.f);
}

__global__ void k_counts(const int* __restrict__ batch, float* __restrict__ c) {
  int n = blockIdx.x * blockDim.x + threadIdx.x;
  if (n < NN) atomicAdd(&c[batch[n]], 1.f);
}

__global__ void k_invs(const float* __restrict__ c, float* __restrict__ s) {
  int g = blockIdx.x * blockDim.x + threadIdx.x;
  if (g < NG) s[g] = rsqrtf(c[g]);
}

__global__ void k_vninit(const float* __restrict__ emb, float* __restrict__ vn) {
  int i = blockIdx.x * blockDim.x + threadIdx.x;
  if (i < NG * NH) vn[i] = emb[i & 63];
}

__global__ void k_fill(float* __restrict__ p, float v, int n) {
  int i = blockIdx.x * blockDim.x + threadIdx.x;
  if (i < n) p[i] = v;
}

// msg = relu(x[row] + edge_attr); agg[col] += msg.  16 lanes/edge, float4 each.
// edge_attr is a 419MB one-shot stream per layer -> non-temporal loads so the
// L2 keeps x (gather source) and agg (atomic target) resident.
__global__ void k_msg(const int* __restrict__ row, const int* __restrict__ col,
                      const float* __restrict__ x, const float* __restrict__ ea,
                      float* __restrict__ agg) {
  long gid = (long)blockIdx.x * blockDim.x + threadIdx.x;
  long e = gid >> 4;
  int part = (int)(gid & 15);
  if (e >= NE) return;
  int r = row[e], c = col[e];
  const v4f xr = *(const v4f*)(x + (long)r * 64 + part * 4);
  const v4f ev = __builtin_nontemporal_load((const v4f*)(ea + e * 64 + part * 4));
  v4f m;
  m.x = fmaxf(xr.x + ev.x, 0.f);
  m.y = fmaxf(xr.y + ev.y, 0.f);
  m.z = fmaxf(xr.z + ev.z, 0.f);
  m.w = fmaxf(xr.w + ev.w, 0.f);
  float* dst = agg + (long)c * 64 + part * 4;
  atomicAdd(dst + 0, m.x);
  atomicAdd(dst + 1, m.y);
  atomicAdd(dst + 2, m.z);
  atomicAdd(dst + 3, m.w);
}

// x = relu(LN(xn * invs[batch])) + h   (in place; h = old x), vnsum += h
__global__ void k_post(float* __restrict__ x, const float* __restrict__ xn,
                       const int* __restrict__ batch,
                       const float* __restrict__ invs,
                       const float* __restrict__ g, const float* __restrict__ b,
                       float* __restrict__ vnsum) {
  int lane = threadIdx.x & 31;
  long n = (long)blockIdx.x * (blockDim.x >> 5) + (threadIdx.x >> 5);
  if (n >= NN) return;
  int gr = batch[n];
  float s = invs[gr];
  long base = n * 64;
  float h1 = x[base + lane], h2 = x[base + lane + 32];
  float v1 = xn[base + lane] * s, v2 = xn[base + lane + 32] * s;
  float mu = wsum32(v1 + v2) * (1.f / 64.f);
  float d1 = v1 - mu, d2 = v2 - mu;
  float var = wsum32(d1 * d1 + d2 * d2) * (1.f / 64.f);
  float r = rsqrtf(var + LN_EPS);
  float y1 = fmaf(d1 * r, g[lane], b[lane]);
  float y2 = fmaf(d2 * r, g[lane + 32], b[lane + 32]);
  x[base + lane] = fmaxf(y1, 0.f) + h1;
  x[base + lane + 32] = fmaxf(y2, 0.f) + h2;
  atomicAdd(&vnsum[(long)gr * 64 + lane], h1);
  atomicAdd(&vnsum[(long)gr * 64 + lane + 32], h2);
}

__global__ void k_lnrelu(const float* __restrict__ in,
                         const float* __restrict__ g,
                         const float* __restrict__ b, float* __restrict__ out,
                         int M) {
  int lane = threadIdx.x & 31;
  int n = blockIdx.x * (blockDim.x >> 5) + (threadIdx.x >> 5);
  if (n >= M) return;
  long base = (long)n * 64;
  float v1 = in[base + lane], v2 = in[base + lane + 32];
  float mu = wsum32(v1 + v2) * (1.f / 64.f);
  float d1 = v1 - mu, d2 = v2 - mu;
  float var = wsum32(d1 * d1 + d2 * d2) * (1.f / 64.f);
  float r = rsqrtf(var + LN_EPS);
  out[base + lane] = fmaxf(fmaf(d1 * r, g[lane], b[lane]), 0.f);
  out[base + lane + 32] =
      fmaxf(fmaf(d2 * r, g[lane + 32], b[lane + 32]), 0.f);
}

__global__ void k_addvn(float* __restrict__ x, const float* __restrict__ vn,
                        const int* __restrict__ batch) {
  long i = (long)blockIdx.x * blockDim.x + threadIdx.x;
  if (i >= (long)NN * 64) return;
  long n = i >> 6;
  x[i] += vn[(long)batch[n] * 64 + (i & 63)];
}

// gate[n] = W2 . relu(BN(hg[n,:])) + b2   (wave per node, 4 cols/lane)
__global__ void k_gate(const float* __restrict__ hg,
                       const float* __restrict__ mean,
                       const float* __restrict__ var,
                       const float* __restrict__ gg,
                       const float* __restrict__ gb,
                       const float* __restrict__ w2,
                       const float* __restrict__ b2, float* __restrict__ gate) {
  int lane = threadIdx.x & 31;
  long n = (long)blockIdx.x * (blockDim.x >> 5) + (threadIdx.x >> 5);
  if (n >= NN) return;
  float acc = 0.f;
#pragma unroll
  for (int q = 0; q < 4; q++) {
    int j = lane * 4 + q;
    float v = hg[n * 128 + j];
    float t = (v - mean[j]) * rsqrtf(var[j] + BN_EPS) * gg[j] + gb[j];
    acc = fmaf(fmaxf(t, 0.f), w2[j], acc);
  }
  acc = wsum32(acc);
  if (lane == 0) gate[n] = acc + b2[0];
}

__global__ void k_max(const float* __restrict__ gate,
                      const int* __restrict__ batch, float* __restrict__ m) {
  int n = blockIdx.x * blockDim.x + threadIdx.x;
  if (n < NN) atomicMaxF(&m[batch[n]], gate[n]);
}

__global__ void k_expsum(const float* __restrict__ gate,
                         const int* __restrict__ batch,
                         const float* __restrict__ m, float* __restrict__ eb,
                         float* __restrict__ sb) {
  int n = blockIdx.x * blockDim.x + threadIdx.x;
  if (n >= NN) return;
  float e = expf(gate[n] - m[batch[n]]);
  eb[n] = e;
  atomicAdd(&sb[batch[n]], e);
}

__global__ void k_pool(const float* __restrict__ x, const float* __restrict__ eb,
                       const float* __restrict__ sb,
                       const int* __restrict__ batch,
                       float* __restrict__ pooled) {
  int lane = threadIdx.x & 31;
  long n = (long)blockIdx.x * (blockDim.x >> 5) + (threadIdx.x >> 5);
  if (n >= NN) return;
  int g = batch[n];
  float w = eb[n] / sb[g];
  atomicAdd(&pooled[(long)g * 64 + lane], w * x[n * 64 + lane]);
  atomicAdd(&pooled[(long)g * 64 + lane + 32], w * x[n * 64 + lane + 32]);
}

__global__ void k_fhid(const float* __restrict__ p, const float* __restrict__ W,
                       const float* __restrict__ b, float* __restrict__ f) {
  int i = blockIdx.x * blockDim.x + threadIdx.x;
  if (i >= NG * 64) return;
  int g = i >> 6, j = i & 63;
  float acc = b[j];
  for (int k = 0; k < 64; k++) acc = fmaf(p[g * 64 + k], W[k * 64 + j], acc);
  f[i] = fmaxf(acc, 0.f);
}

__global__ void k_fout(const float* __restrict__ f, const float* __restrict__ W,
                       const float* __restrict__ b, float* __restrict__ out) {
  int lane = threadIdx.x & 31;
  int g = blockIdx.x * (blockDim.x >> 5) + (threadIdx.x >> 5);
  if (g >= NG) return;
  float acc = fmaf(f[g * 64 + lane], W[lane], 0.f) +
              f[g * 64 + lane + 32] * W[lane + 32];
  acc = wsum32(acc);
  if (lane == 0) out[g] = 1.f / (1.f + expf(-(acc + b[0])));
}

// ---------------------------------------------------------------------------
extern "C" void kernel_launch(void* const* d_in, const int* in_sizes, int n_in,
                              void* d_out, int out_size, void* d_ws,
                              size_t ws_size, hipStream_t stream) {
  (void)in_sizes; (void)n_in; (void)out_size; (void)ws_size;
  const float* cfg = (const float*)d_in[0];
  const int* ei = (const int*)d_in[1];
  const int* row = ei;
  const int* col = ei + NE;
  const int* batch = (const int*)d_in[2];
  auto F = [&](int i) { return (const float*)d_in[i]; };
  // param leaf order (dict order): node_mlp 3..6, edge_mlp 7..10,
  // convs[i] 11+5i (l1W,l1b,l2W,l2b,eps), lns[i] 26+2i (g,b),
  // vn_mlps[i] 32+8i (l1W,l1b,ln1g,ln1b,l2W,l2b,ln2g,ln2b),
  // vn_emb 56, gate 57..64 (l1W,l1b,bn_g,bn_b,bn_mean,bn_var,l2W,l2b),
  // mlp 65..68 (l1W,l1b,l2W,l2b)

  float* ws = (float*)d_ws;
  size_t off = 0;
  auto alloc = [&](size_t nf) { float* p = ws + off; off += nf; return p; };
  float* deg = alloc(NN);
  float* x = alloc((size_t)NN * 64);
  float* t1 = alloc((size_t)NN * 64);
  float* t2 = alloc((size_t)NN * 64);
  float* agg = alloc((size_t)NN * 64);
  float* gbuf = alloc((size_t)NN * 128);
  float* gate = alloc(NN);
  float* ebuf = alloc(NN);
  float* vn = alloc(NG * 64);
  float* vns = alloc(NG * 64);
  float* g1 = alloc(NG * 64);
  float* g2 = alloc(NG * 64);
  float* cnt = alloc(NG);
  float* invs = alloc(NG);
  float* mb = alloc(NG);
  float* sb = alloc(NG);
  float* pooled = alloc(NG * 64);
  float* fb = alloc(NG * 64);
  float* ea = alloc((size_t)NE * 64);  // 419 MB, largest last

  hipMemsetAsync(deg, 0, NN * sizeof(float), stream);
  hipMemsetAsync(cnt, 0, NG * sizeof(float), stream);
  k_deg<<<(NE + 255) / 256, 256, 0, stream>>>(col, deg);
  k_counts<<<(NN + 255) / 256, 256, 0, stream>>>(batch, cnt);
  k_invs<<<1, 256, 0, stream>>>(cnt, invs);

  // x = node_mlp(config): l1 fused into WMMA A-producer
  gemm_wmma<1, 64, false, false><<<NN / 64, 512, 0, stream>>>(
      nullptr, cfg, nullptr, nullptr, F(5), F(6), F(3), F(4), nullptr, x);
  // edge_attr = edge_mlp([deg[row],deg[col]]): l1 fused, NT stores
  gemm_wmma<2, 64, false, true><<<NE / 64, 512, 0, stream>>>(
      nullptr, deg, row, col, F(9), F(10), F(7), F(8), nullptr, ea);
  k_vninit<<<(NG * 64 + 255) / 256, 256, 0, stream>>>(F(56), vn);

  for (int i = 0; i < GDEPTH; i++) {
    const int cb = 11 + 5 * i, lb = 26 + 2 * i, vb = 32 + 8 * i;
    hipMemsetAsync(agg, 0, (size_t)NN * 64 * sizeof(float), stream);
    hipMemsetAsync(vns, 0, NG * 64 * sizeof(float), stream);
    k_msg<<<(int)(((size_t)NE * 16) / 256), 256, 0, stream>>>(row, col, x, ea,
                                                              agg);
    // t1 = relu( ((1+eps)x + agg) @ l1W + l1b )
    gemm_wmma<3, 64, true, false><<<NN / 64, 512, 0, stream>>>(
        x, agg, nullptr, nullptr, F(cb + 0), F(cb + 1), nullptr, nullptr,
        F(cb + 4), t1);
    // t2 = t1 @ l2W + l2b
    gemm_wmma<0, 64, false, false><<<NN / 64, 512, 0, stream>>>(
        t1, nullptr, nullptr, nullptr, F(cb + 2), F(cb + 3), nullptr, nullptr,
        nullptr, t2);
    k_post<<<NN / 8, 256, 0, stream>>>(x, t2, batch, invs, F(lb), F(lb + 1),
                                       vns);
    // virtual node: g1 = (vns + vn) @ q.l1 + b
    gemm_wmma<3, 64, false, false><<<NG / 64, 512, 0, stream>>>(
        vns, vn, nullptr, nullptr, F(vb + 0), F(vb + 1), nullptr, nullptr,
        nullptr, g1);
    k_lnrelu<<<NG / 8, 256, 0, stream>>>(g1, F(vb + 2), F(vb + 3), g2, NG);
    gemm_wmma<0, 64, false, false><<<NG / 64, 512, 0, stream>>>(
        g2, nullptr, nullptr, nullptr, F(vb + 4), F(vb + 5), nullptr, nullptr,
        nullptr, g1);
    k_lnrelu<<<NG / 8, 256, 0, stream>>>(g1, F(vb + 6), F(vb + 7), vn, NG);
    k_addvn<<<(int)(((size_t)NN * 64) / 256), 256, 0, stream>>>(x, vn, batch);
  }

  // gate hidden: gbuf = x @ gate.l1W + b  (NOUT=128)
  gemm_wmma<0, 128, false, false><<<NN / 64, 1024, 0, stream>>>(
      x, nullptr, nullptr, nullptr, F(57), F(58), nullptr, nullptr, nullptr,
      gbuf);
  k_gate<<<NN / 8, 256, 0, stream>>>(gbuf, F(61), F(62), F(59), F(60), F(63),
                                     F(64), gate);
  k_fill<<<1, 256, 0, stream>>>(mb, -FLT_MAX, NG);
  hipMemsetAsync(sb, 0, NG * sizeof(float), stream);
  hipMemsetAsync(pooled, 0, NG * 64 * sizeof(float), stream);
  k_max<<<(NN + 255) / 256, 256, 0, stream>>>(gate, batch, mb);
  k_expsum<<<(NN + 255) / 256, 256, 0, stream>>>(gate, batch, mb, ebuf, sb);
  k_pool<<<NN / 8, 256, 0, stream>>>(x, ebuf, sb, batch, pooled);
  k_fhid<<<(NG * 64 + 255) / 256, 256, 0, stream>>>(pooled, F(65), F(66), fb);
  k_fout<<<NG / 8, 256, 0, stream>>>(fb, F(67), F(68), (float*)d_out);
}